// GAT_24300924961368
// MI455X (gfx1250) — compile-verified
//
#include <hip/hip_runtime.h>
#include <math.h>

typedef float v2f __attribute__((ext_vector_type(2)));
typedef float v8f __attribute__((ext_vector_type(8)));

// ---------- helpers ----------
__device__ __forceinline__ unsigned flipf(float f) {
    unsigned u = __float_as_uint(f);
    return u ^ ((unsigned)((int)u >> 31) | 0x80000000u);
}
__device__ __forceinline__ float unflipf(unsigned u) {
    unsigned mask = ((int)u < 0) ? 0x80000000u : 0xFFFFFFFFu;
    return __uint_as_float(u ^ mask);
}
__device__ __forceinline__ float leaky(float v) { return v > 0.f ? v : 0.2f * v; }

// ---------- WMMA fp32 GEMM: C[M,NCOL] = A[M,K] * B[K,NCOL], M % 16 == 0 ----------
// One wave computes NT adjacent 16x16 output tiles (16 x 16*NT strip), reusing each
// A fetch across NT V_WMMA_F32_16X16X4_F32 issues. K/NCOL are compile-time so all
// loads are immediate-offset global_load's; out-of-range columns are clamped on
// load (no EXEC divergence in the loop) and masked only at the store.
template<int K, int NCOL, int NT>
__global__ __launch_bounds__(256) void wmma_gemm_f32(
    const float* __restrict__ A, const float* __restrict__ B, float* __restrict__ C,
    int M)
{
    constexpr int GROUPS_N = (NCOL + 16 * NT - 1) / (16 * NT);
    int wid  = (blockIdx.x * blockDim.x + threadIdx.x) >> 5;   // wave-uniform
    int lane = threadIdx.x & 31;
    int tilesM = M >> 4;
    int tm = wid / GROUPS_N;
    int tg = wid - tm * GROUPS_N;
    if (tm >= tilesM) return;                                   // wave-uniform exit

    int rowA  = tm * 16 + (lane & 15);
    int khalf = (lane >> 4) << 1;                               // 0 or 2
    const float* Arow = A + (size_t)rowA * K + khalf;

    int col[NT];
    const float* Bcol[NT];
#pragma unroll
    for (int t = 0; t < NT; ++t) {
        col[t] = (tg * NT + t) * 16 + (lane & 15);
        int cc = col[t] < NCOL ? col[t] : NCOL - 1;             // clamp: loads stay in bounds
        Bcol[t] = B + (size_t)khalf * NCOL + cc;
    }

    v8f acc[NT] = {};

#pragma unroll 8
    for (int k = 0; k < K; k += 4) {
        // A 16x4: lane<16 holds K=k+0,k+1; lane>=16 holds K=k+2,k+3 (contiguous pair)
        float2 av = *(const float2*)(Arow + k);
        v2f a; a.x = av.x; a.y = av.y;
#pragma unroll
        for (int t = 0; t < NT; ++t) {
            // B 4x16: lane<16 holds K=k+0,k+1 at N=col; lane>=16 holds K=k+2,k+3
            v2f b;
            b.x = Bcol[t][(size_t)k * NCOL];
            b.y = Bcol[t][(size_t)k * NCOL + NCOL];
            acc[t] = __builtin_amdgcn_wmma_f32_16x16x4_f32(false, a, false, b,
                                                           (short)0, acc[t], false, false);
        }
    }

    // C/D: VGPR r -> row r (lanes 0-15) or r+8 (lanes 16-31), col = lane&15
    int rbase = tm * 16 + ((lane >> 4) << 3);
#pragma unroll
    for (int t = 0; t < NT; ++t) {
        if (col[t] < NCOL) {
            float* Crow = C + (size_t)rbase * NCOL + col[t];
#pragma unroll
            for (int r = 0; r < 8; ++r)
                Crow[(size_t)r * NCOL] = acc[t][r];
        }
    }
}

// ---------- per-node attention coefficients ----------
// a_src[n,h] = sum_c h[n,h,c]*att_src[h,c];  a_dst likewise. One thread per (n,h).
__global__ void attn_coef(const float* __restrict__ h,
                          const float* __restrict__ att_src,
                          const float* __restrict__ att_dst,
                          float* __restrict__ a_src, float* __restrict__ a_dst,
                          int N, int H, int C)
{
    int tid = blockIdx.x * blockDim.x + threadIdx.x;
    if (tid >= N * H) return;
    int h_ = tid % H;
    const float4* hv = (const float4*)(h + (size_t)tid * C);
    const float4* as = (const float4*)(att_src + (size_t)h_ * C);
    const float4* ad = (const float4*)(att_dst + (size_t)h_ * C);
    float s = 0.f, d = 0.f;
    for (int c = 0; c < (C >> 2); ++c) {
        float4 hh = hv[c], sa = as[c], da = ad[c];
        s += hh.x * sa.x + hh.y * sa.y + hh.z * sa.z + hh.w * sa.w;
        d += hh.x * da.x + hh.y * da.y + hh.z * da.z + hh.w * da.w;
    }
    a_src[tid] = s;
    a_dst[tid] = d;
}

// ---------- segment-max over incoming edges (flipped-uint atomicMax) ----------
__global__ void edge_max_k(const int* __restrict__ esrc, const int* __restrict__ edst,
                           const float* __restrict__ a_src, const float* __restrict__ a_dst,
                           unsigned* __restrict__ emax, int E, int N, int H)
{
    int tid = blockIdx.x * blockDim.x + threadIdx.x;
    int total = (E + N) * H;
    if (tid >= total) return;
    int e = tid / H, h_ = tid - e * H;
    int s, d;
    if (e < E) { s = esrc[e]; d = edst[e]; } else { s = d = e - E; }   // self-loops
    float v = leaky(a_src[s * H + h_] + a_dst[d * H + h_]);
    atomicMax(&emax[d * H + h_], flipf(v));
}

// ---------- segment-sum of exp(e - max) ----------
__global__ void edge_sum_k(const int* __restrict__ esrc, const int* __restrict__ edst,
                           const float* __restrict__ a_src, const float* __restrict__ a_dst,
                           const unsigned* __restrict__ emax, float* __restrict__ denom,
                           int E, int N, int H)
{
    int tid = blockIdx.x * blockDim.x + threadIdx.x;
    int total = (E + N) * H;
    if (tid >= total) return;
    int e = tid / H, h_ = tid - e * H;
    int s, d;
    if (e < E) { s = esrc[e]; d = edst[e]; } else { s = d = e - E; }
    float v = leaky(a_src[s * H + h_] + a_dst[d * H + h_]);
    float m = unflipf(emax[d * H + h_]);
    atomicAdd(&denom[d * H + h_], __expf(v - m));
}

// ---------- weighted aggregation: out[dst] += alpha * h[src] ----------
// One wave per edge; lane-strided coalesced chunks; f32 atomics into L2-resident out.
__global__ __launch_bounds__(256) void edge_agg_k(
    const int* __restrict__ esrc, const int* __restrict__ edst,
    const float* __restrict__ h,
    const float* __restrict__ a_src, const float* __restrict__ a_dst,
    const unsigned* __restrict__ emax, const float* __restrict__ denom,
    float* __restrict__ out, int E, int N, int H, int C)
{
    int wid  = (blockIdx.x * blockDim.x + threadIdx.x) >> 5;
    int lane = threadIdx.x & 31;
    if (wid >= E + N) return;
    int s, d;
    if (wid < E) { s = esrc[wid]; d = edst[wid]; } else { s = d = wid - E; }
    int HC = H * C;
    const float* hs = h   + (size_t)s * HC;
    float*       od = out + (size_t)d * HC;
    for (int off = lane; off < HC; off += 32) {
        int h_ = off / C;
        float v = leaky(a_src[s * H + h_] + a_dst[d * H + h_]);
        float m = unflipf(emax[d * H + h_]);
        float alpha = __expf(v - m) / denom[d * H + h_];
        atomicAdd(&od[off], hs[off] * alpha);
    }
}

// ---------- bias + ELU (layer 1 epilogue) ----------
__global__ void bias_elu_k(float* __restrict__ x, const float* __restrict__ b,
                           int total, int F)
{
    int tid = blockIdx.x * blockDim.x + threadIdx.x;
    if (tid >= total) return;
    float v = x[tid] + b[tid % F];
    x[tid] = v > 0.f ? v : __expf(v) - 1.f;
}

// ---------- bias only (final output) ----------
__global__ void bias_out_k(const float* __restrict__ x, const float* __restrict__ b,
                           float* __restrict__ out, int total, int F)
{
    int tid = blockIdx.x * blockDim.x + threadIdx.x;
    if (tid >= total) return;
    out[tid] = x[tid] + b[tid % F];
}

// =================================================================================
extern "C" void kernel_launch(void* const* d_in, const int* in_sizes, int n_in,
                              void* d_out, int out_size, void* d_ws, size_t ws_size,
                              hipStream_t stream)
{
    const float* x        = (const float*)d_in[0];
    const int*   edge     = (const int*)d_in[1];   // [2, E] int32
    const float* W1       = (const float*)d_in[2];
    const float* att_src1 = (const float*)d_in[3];
    const float* att_dst1 = (const float*)d_in[4];
    const float* b1       = (const float*)d_in[5];
    const float* W2       = (const float*)d_in[6];
    const float* att_src2 = (const float*)d_in[7];
    const float* att_dst2 = (const float*)d_in[8];
    const float* b2       = (const float*)d_in[9];
    float* outp = (float*)d_out;

    const int F_IN = 256, H1 = 4, C1 = 64, HC1 = 256, NCLS = 40;
    const int N = in_sizes[0] / F_IN;      // 50000 (multiple of 16)
    const int E = in_sizes[1] / 2;         // 1600000
    const int* esrc = edge;
    const int* edst = edge + E;

    // --- bump allocator over workspace ---
    char* wp = (char*)d_ws;
    auto alloc = [&](size_t bytes) -> void* {
        void* p = wp; wp += (bytes + 255) & ~(size_t)255; return p;
    };
    float*    h1  = (float*)   alloc((size_t)N * HC1 * 4);
    float*    as1 = (float*)   alloc((size_t)N * H1 * 4);
    float*    ad1 = (float*)   alloc((size_t)N * H1 * 4);
    unsigned* em1 = (unsigned*)alloc((size_t)N * H1 * 4);
    float*    dn1 = (float*)   alloc((size_t)N * H1 * 4);
    float*    o1  = (float*)   alloc((size_t)N * HC1 * 4);
    float*    h2  = (float*)   alloc((size_t)N * NCLS * 4);
    float*    as2 = (float*)   alloc((size_t)N * 4);
    float*    ad2 = (float*)   alloc((size_t)N * 4);
    unsigned* em2 = (unsigned*)alloc((size_t)N * 4);
    float*    dn2 = (float*)   alloc((size_t)N * 4);
    float*    o2  = (float*)   alloc((size_t)N * NCLS * 4);

    // accumulator init (flip(-inf) > 0, every node has a self-loop, so 0 is safe)
    hipMemsetAsync(em1, 0, (size_t)N * H1 * 4, stream);
    hipMemsetAsync(dn1, 0, (size_t)N * H1 * 4, stream);
    hipMemsetAsync(o1,  0, (size_t)N * HC1 * 4, stream);
    hipMemsetAsync(em2, 0, (size_t)N * 4, stream);
    hipMemsetAsync(dn2, 0, (size_t)N * 4, stream);
    hipMemsetAsync(o2,  0, (size_t)N * NCLS * 4, stream);

    const int BS = 256;
    const long ET = (long)E + N;

    // ================= Layer 1 =================
    {   // h1 = x @ W1   (fp32 WMMA, 4 column tiles per wave)
        const int GROUPS_N = 256 / (16 * 4);                 // 4
        long waves = (long)(N / 16) * GROUPS_N;
        long thr = waves * 32;
        wmma_gemm_f32<256, 256, 4><<<(thr + BS - 1) / BS, BS, 0, stream>>>(x, W1, h1, N);
    }
    {   long t = (long)N * H1;
        attn_coef<<<(t + BS - 1) / BS, BS, 0, stream>>>(h1, att_src1, att_dst1, as1, ad1, N, H1, C1);
    }
    {   long t = ET * H1;
        edge_max_k<<<(t + BS - 1) / BS, BS, 0, stream>>>(esrc, edst, as1, ad1, em1, E, N, H1);
        edge_sum_k<<<(t + BS - 1) / BS, BS, 0, stream>>>(esrc, edst, as1, ad1, em1, dn1, E, N, H1);
    }
    {   long thr = ET * 32;
        edge_agg_k<<<(thr + BS - 1) / BS, BS, 0, stream>>>(esrc, edst, h1, as1, ad1, em1, dn1,
                                                           o1, E, N, H1, C1);
    }
    {   long t = (long)N * HC1;
        bias_elu_k<<<(t + BS - 1) / BS, BS, 0, stream>>>(o1, b1, (int)t, HC1);
    }

    // ================= Layer 2 =================
    {   // h2 = o1 @ W2  (40 cols -> one wave covers all 3 column tiles, store-masked)
        long waves = (long)(N / 16);                         // GROUPS_N == 1
        long thr = waves * 32;
        wmma_gemm_f32<256, 40, 3><<<(thr + BS - 1) / BS, BS, 0, stream>>>(o1, W2, h2, N);
    }
    {   long t = (long)N;
        attn_coef<<<(t + BS - 1) / BS, BS, 0, stream>>>(h2, att_src2, att_dst2, as2, ad2, N, 1, NCLS);
    }
    {   long t = ET;
        edge_max_k<<<(t + BS - 1) / BS, BS, 0, stream>>>(esrc, edst, as2, ad2, em2, E, N, 1);
        edge_sum_k<<<(t + BS - 1) / BS, BS, 0, stream>>>(esrc, edst, as2, ad2, em2, dn2, E, N, 1);
    }
    {   long thr = ET * 32;
        edge_agg_k<<<(thr + BS - 1) / BS, BS, 0, stream>>>(esrc, edst, h2, as2, ad2, em2, dn2,
                                                           o2, E, N, 1, NCLS);
    }
    {   long t = (long)N * NCLS;   // heads==1 -> mean over heads is identity
        bias_out_k<<<(t + BS - 1) / BS, BS, 0, stream>>>(o2, b2, outp, (int)t, NCLS);
    }
}